// GNNActor_90701119357780
// MI455X (gfx1250) — compile-verified
//
#include <hip/hip_runtime.h>
#include <hip/hip_bf16.h>

typedef __attribute__((ext_vector_type(2))) float v2f;
typedef __attribute__((ext_vector_type(8))) float v8f;

// ---------------- degree / normalization ----------------

__global__ void k_init_deg(float* __restrict__ deg, int n) {
    int i = blockIdx.x * blockDim.x + threadIdx.x;
    if (i < n) deg[i] = 1.0f;   // self-loop contributes 1 to every node's degree
}

__global__ void k_count_deg(const int* __restrict__ ei, float* __restrict__ deg, int nedges) {
    int e = blockIdx.x * blockDim.x + threadIdx.x;
    if (e < nedges) atomicAdd(&deg[ei[nedges + e]], 1.0f);   // dst = ei[E + e]
}

__global__ void k_rsqrt_inplace(float* __restrict__ deg, int n) {
    int i = blockIdx.x * blockDim.x + threadIdx.x;
    if (i < n) deg[i] = rsqrtf(deg[i]);
}

// w[e] = dinv[src]*dinv[dst]  (precomputed once, reused by both GCN layers)
__global__ void k_edge_norm(const int* __restrict__ ei, const float* __restrict__ dinv,
                            float* __restrict__ w, int nedges) {
    int e = blockIdx.x * blockDim.x + threadIdx.x;
    if (e < nedges) w[e] = dinv[ei[e]] * dinv[ei[nedges + e]];
}

// ---------------- dense 64-wide GEMM via V_WMMA_F32_16X16X4_F32 ----------------
// C[nrows x 64] = A[nrows x 64] * W[64 x 64]  (+ bias, relu if BIAS_RELU)
// W staged once per block into LDS, transposed with pitch 68 so each B fragment
// (k0, k0+1 at fixed column) is one bank-conflict-free ds_load_b64.

#define WT_PITCH 68   // 64 + 4 pad: dword bank = (4*col + k) % 64 -> conflict-free b64

template <bool BIAS_RELU>
__global__ void k_gemm64(const float* __restrict__ A, const float* __restrict__ W,
                         const float* __restrict__ bias, float* __restrict__ C, int nrows) {
    __shared__ float ldsW[64 * WT_PITCH];     // Wt[col][k], 17.4 KB of the 320 KB WGP LDS

    // stage W (4096 floats) cooperatively: coalesced global reads, transposed LDS writes
    for (int idx = threadIdx.x; idx < 4096; idx += 256) {
        const int k = idx >> 6;
        const int col = idx & 63;
        ldsW[col * WT_PITCH + k] = W[idx];
    }
    __syncthreads();

    const int lane = threadIdx.x & 31;
    const int wave = threadIdx.x >> 5;
    const int rowTiles = (nrows + 15) >> 4;
    const int tile = blockIdx.x * 8 + wave;       // 8 waves per 256-thread block
    if (tile < rowTiles) {                        // wave-uniform: EXEC all-ones inside
        const int half = lane >> 4;               // 0: lanes 0-15, 1: lanes 16-31
        const int m    = lane & 15;
        int row = tile * 16 + m;
        if (row >= nrows) row = nrows - 1;        // clamped duplicate read (tail safety)
        const float* __restrict__ arow = A + (size_t)row * 64;

        v8f c0 = {}, c1 = {}, c2 = {}, c3 = {};
#pragma unroll
        for (int kk = 0; kk < 16; ++kk) {
            const int k0 = kk * 4 + 2 * half;     // A vgpr j holds K = 2*half + j
            v2f a;
            a.x = arow[k0];
            a.y = arow[k0 + 1];
            // B vgpr j: K = 2*half + j, col = nt*16 + m  -> one b64 from transposed LDS
            v2f b0 = *(const v2f*)&ldsW[(0 * 16 + m) * WT_PITCH + k0];
            v2f b1 = *(const v2f*)&ldsW[(1 * 16 + m) * WT_PITCH + k0];
            v2f b2 = *(const v2f*)&ldsW[(2 * 16 + m) * WT_PITCH + k0];
            v2f b3 = *(const v2f*)&ldsW[(3 * 16 + m) * WT_PITCH + k0];
            c0 = __builtin_amdgcn_wmma_f32_16x16x4_f32(false, a, false, b0, (short)0, c0, false, false);
            c1 = __builtin_amdgcn_wmma_f32_16x16x4_f32(false, a, false, b1, (short)0, c1, false, false);
            c2 = __builtin_amdgcn_wmma_f32_16x16x4_f32(false, a, false, b2, (short)0, c2, false, false);
            c3 = __builtin_amdgcn_wmma_f32_16x16x4_f32(false, a, false, b3, (short)0, c3, false, false);
        }

        v8f acc[4] = {c0, c1, c2, c3};
#pragma unroll
        for (int nt = 0; nt < 4; ++nt) {
            const int col = nt * 16 + m;
            const float bv = BIAS_RELU ? bias[col] : 0.0f;
#pragma unroll
            for (int r = 0; r < 8; ++r) {
                const int orow = tile * 16 + r + 8 * half;   // C vgpr r -> M = r + 8*half
                if (orow < nrows) {
                    float v = acc[nt][r];
                    if (BIAS_RELU) v = fmaxf(v + bv, 0.0f);
                    C[(size_t)orow * 64 + col] = v;
                }
            }
        }
    }
}

// ---------------- GCN aggregation ----------------

// agg[i,:] = dinv[i]^2 * t[i,:]   (self-loop term; also initializes poisoned ws)
__global__ void k_selfloop_init(const float* __restrict__ t, const float* __restrict__ dinv,
                                float* __restrict__ agg, int n) {
    long tid = (long)blockIdx.x * blockDim.x + threadIdx.x;
    int i = (int)(tid >> 4);
    int j = (int)(tid & 15) * 4;
    if (i >= n) return;
    float w = dinv[i] * dinv[i];
    float4 v = *(const float4*)(t + (size_t)i * 64 + j);
    float4 r = make_float4(w * v.x, w * v.y, w * v.z, w * v.w);
    *(float4*)(agg + (size_t)i * 64 + j) = r;
}

// agg[dst,:] += w[e] * t[src,:]   (16 threads per edge, float4 each)
__global__ void k_edge_scatter(const int* __restrict__ ei, const float* __restrict__ wnorm,
                               const float* __restrict__ t, float* __restrict__ agg, int nedges) {
    long tid = (long)blockIdx.x * blockDim.x + threadIdx.x;
    int e = (int)(tid >> 4);
    int j = (int)(tid & 15) * 4;
    if (e >= nedges) return;
    int s = ei[e];
    int d = ei[nedges + e];
    float w = wnorm[e];
    float4 v = *(const float4*)(t + (size_t)s * 64 + j);
    float* p = agg + (size_t)d * 64 + j;
    atomicAdd(p + 0, w * v.x);
    atomicAdd(p + 1, w * v.y);
    atomicAdd(p + 2, w * v.z);
    atomicAdd(p + 3, w * v.w);
}

// h = relu(agg + b)
__global__ void k_bias_relu(const float* __restrict__ agg, const float* __restrict__ b,
                            float* __restrict__ h, long n64) {
    long idx = (long)blockIdx.x * blockDim.x + threadIdx.x;
    if (idx >= n64) return;
    int f = (int)(idx & 63);
    h[idx] = fmaxf(agg[idx] + b[f], 0.0f);
}

// out[i] = h[i,:] . w + b0   (no relu on final layer)
__global__ void k_final_dot(const float* __restrict__ H, const float* __restrict__ w,
                            const float* __restrict__ b, float* __restrict__ out, int n) {
    int i = blockIdx.x * blockDim.x + threadIdx.x;
    if (i >= n) return;
    const float4* hp = (const float4*)(H + (size_t)i * 64);
    const float4* wp = (const float4*)w;
    float s = 0.0f;
#pragma unroll
    for (int k = 0; k < 16; ++k) {
        float4 hv = hp[k], wv = wp[k];
        s += hv.x * wv.x + hv.y * wv.y + hv.z * wv.z + hv.w * wv.w;
    }
    out[i] = s + b[0];
}

// ---------------- driver ----------------

extern "C" void kernel_launch(void* const* d_in, const int* in_sizes, int n_in,
                              void* d_out, int out_size, void* d_ws, size_t ws_size,
                              hipStream_t stream) {
    (void)n_in; (void)out_size; (void)ws_size;

    const float* x   = (const float*)d_in[0];
    const int*   ei  = (const int*)  d_in[1];
    const float* W1  = (const float*)d_in[2];
    const float* b1  = (const float*)d_in[3];
    const float* W2  = (const float*)d_in[4];
    const float* b2  = (const float*)d_in[5];
    const float* Wf1 = (const float*)d_in[6];
    const float* bf1 = (const float*)d_in[7];
    const float* Wf2 = (const float*)d_in[8];
    const float* bf2 = (const float*)d_in[9];
    const float* Wf3 = (const float*)d_in[10];
    const float* bf3 = (const float*)d_in[11];
    float* out = (float*)d_out;

    const int n = in_sizes[0] / 64;   // 100000
    const int E = in_sizes[1] / 2;    // 1600000
    const long n64 = (long)n * 64;

    float* ws    = (float*)d_ws;
    float* dinv  = ws;                          // [n]
    float* wnorm = ws + n;                      // [E]
    float* bufA  = wnorm + E;                   // [n*64]
    float* bufB  = bufA + (size_t)n * 64;       // [n*64]

    const int T = 256;
    const int rowTiles = (n + 15) / 16;
    const int gemmBlocks = (rowTiles + 7) / 8;
    const int nodeVecBlocks = (int)(((long)n * 16 + T - 1) / T);
    const int edgeVecBlocks = (int)(((long)E * 16 + T - 1) / T);
    const int elemBlocks = (int)((n64 + T - 1) / T);

    // normalization: deg -> dinv -> per-edge weights
    k_init_deg<<<(n + T - 1) / T, T, 0, stream>>>(dinv, n);
    k_count_deg<<<(E + T - 1) / T, T, 0, stream>>>(ei, dinv, E);
    k_rsqrt_inplace<<<(n + T - 1) / T, T, 0, stream>>>(dinv, n);
    k_edge_norm<<<(E + T - 1) / T, T, 0, stream>>>(ei, dinv, wnorm, E);

    // GCN layer 1: t1 = x @ W1 -> bufA; agg -> bufB; h1 = relu(agg+b1) -> bufA
    k_gemm64<false><<<gemmBlocks, T, 0, stream>>>(x, W1, nullptr, bufA, n);
    k_selfloop_init<<<nodeVecBlocks, T, 0, stream>>>(bufA, dinv, bufB, n);
    k_edge_scatter<<<edgeVecBlocks, T, 0, stream>>>(ei, wnorm, bufA, bufB, E);
    k_bias_relu<<<elemBlocks, T, 0, stream>>>(bufB, b1, bufA, n64);

    // GCN layer 2: t2 = h1 @ W2 -> bufB; agg -> bufA; h2 = relu(agg+b2) -> bufB
    k_gemm64<false><<<gemmBlocks, T, 0, stream>>>(bufA, W2, nullptr, bufB, n);
    k_selfloop_init<<<nodeVecBlocks, T, 0, stream>>>(bufB, dinv, bufA, n);
    k_edge_scatter<<<edgeVecBlocks, T, 0, stream>>>(ei, wnorm, bufB, bufA, E);
    k_bias_relu<<<elemBlocks, T, 0, stream>>>(bufA, b2, bufB, n64);

    // MLP head (bias+relu fused into WMMA epilogue)
    k_gemm64<true><<<gemmBlocks, T, 0, stream>>>(bufB, Wf1, bf1, bufA, n);
    k_gemm64<true><<<gemmBlocks, T, 0, stream>>>(bufA, Wf2, bf2, bufB, n);
    k_final_dot<<<(n + T - 1) / T, T, 0, stream>>>(bufB, Wf3, bf3, out, n);
}